// ModelPipeline_74766790688909
// MI455X (gfx1250) — compile-verified
//
#include <hip/hip_runtime.h>

// Sizes from the reference
#define Hn     1024
#define GH     4096          // 4*H gate rows
#define OUTN   512
#define DEPTH  32
#define EPSV   1e-8f

#define WAVES  4             // waves per gemv block (128 threads)
#define ROWS_PER_BLOCK (WAVES * 16)
#define KSPLIT 4
#define KCHUNK (Hn / KSPLIT) // 256

typedef __attribute__((ext_vector_type(2))) float v2f;
typedef __attribute__((ext_vector_type(8))) float v8f;

// ---------------------------------------------------------------------------
// GEMV via V_WMMA_F32_16X16X4_F32.
// Each wave owns 16 rows of W ([rows x 1024], row-major). Per K-chunk of 4:
//   A (16x4 f32): lane L<16 -> row L,   elems W[row][k+0..1] (VGPR0/1)
//                 lane L>=16 -> row L-16, elems W[row][k+2..3]
//   -> one contiguous b64 load per lane.
//   B (4x16 f32): x[k..k+3] broadcast to all 16 columns (LDS broadcast).
// All columns of D are then identical; column 0 sits in lane 0 (M=0..7,
// acc[0..7]) and lane 16 (M=8..15).
// grid = (rows/64, nMat{1|2}, KSPLIT); blockIdx.y picks (W1,x1)/(W2,x2);
// blockIdx.z picks the K slice. Partial sums go to g[(y*KSPLIT+z)*GH + row].
// ---------------------------------------------------------------------------
__global__ __launch_bounds__(WAVES * 32)
void gemv_wmma(const float* __restrict__ W1, const float* __restrict__ x1,
               const float* __restrict__ W2, const float* __restrict__ x2,
               float* __restrict__ g)
{
    const float* W  = blockIdx.y ? W2 : W1;
    const float* xv = blockIdx.y ? x2 : x1;

    __shared__ float sx[KCHUNK];
    const int tid   = threadIdx.x;
    const int kBase = blockIdx.z * KCHUNK;
    for (int i = tid; i < KCHUNK; i += WAVES * 32) sx[i] = xv[kBase + i];
    __syncthreads();

    const int wave    = tid >> 5;
    const int lane    = tid & 31;
    const int khalf   = lane >> 4;                       // 0: K0/K1, 1: K2/K3
    const int rowBase = (blockIdx.x * WAVES + wave) * 16;
    const int row     = rowBase + (lane & 15);
    const float* wrow = W + (size_t)row * Hn + kBase + 2 * khalf;
    const float* xs   = sx + 2 * khalf;

    v8f acc = {0.f, 0.f, 0.f, 0.f, 0.f, 0.f, 0.f, 0.f};
#pragma unroll 8
    for (int k = 0; k < KCHUNK; k += 4) {
        v2f a = *(const v2f*)(wrow + k);   // W[row][kBase+k+2*khalf +{0,1}]
        v2f b;
        b.x = xs[k];                       // x[kBase+k+2*khalf]
        b.y = xs[k + 1];
        acc = __builtin_amdgcn_wmma_f32_16x16x4_f32(
            false, a, false, b, (short)0, acc, false, false);
    }

    float* out = g + (size_t)(blockIdx.y * KSPLIT + blockIdx.z) * GH + rowBase;
    if (lane == 0) {
#pragma unroll
        for (int v = 0; v < 8; ++v) out[v] = acc[v];       // rows rowBase+0..7
    } else if (lane == 16) {
#pragma unroll
        for (int v = 0; v < 8; ++v) out[8 + v] = acc[v];   // rows rowBase+8..15
    }
}

__device__ __forceinline__ float fsigmoid(float x) {
    return 1.0f / (1.0f + __expf(-x));
}
__device__ __forceinline__ float ftanh_fast(float x) {
    float c = fminf(fmaxf(x, -15.0f), 15.0f);
    float e = __expf(2.0f * c);
    return (e - 1.0f) / (e + 1.0f);
}

// ---------------------------------------------------------------------------
// One LSTM cell: sum 2*KSPLIT gate partials + biases, gate nonlinearity,
// h/c update, optional L2-normalize into x_next. Single 1024-thread block.
// ---------------------------------------------------------------------------
__global__ __launch_bounds__(Hn)
void lstm_cell(const float* __restrict__ g,
               const float* __restrict__ bih, const float* __restrict__ bhh,
               const float* __restrict__ c_prev,
               float* __restrict__ h_out, float* __restrict__ c_out,
               float* __restrict__ x_next, int do_norm)
{
    __shared__ float red[Hn];
    const int j = threadIdx.x;

    float gate[4];
#pragma unroll
    for (int q = 0; q < 4; ++q) {
        const int r = q * Hn + j;
        float s = bih[r] + bhh[r];
#pragma unroll
        for (int p = 0; p < 2 * KSPLIT; ++p) s += g[(size_t)p * GH + r];
        gate[q] = s;
    }

    const float i  = fsigmoid(gate[0]);
    const float f  = fsigmoid(gate[1]);
    const float gg = ftanh_fast(gate[2]);
    const float o  = fsigmoid(gate[3]);
    const float c2 = f * c_prev[j] + i * gg;
    const float h  = o * ftanh_fast(c2);

    h_out[j] = h;
    c_out[j] = c2;

    if (do_norm) {
        red[j] = h * h;
        __syncthreads();
        for (int s = Hn / 2; s > 0; s >>= 1) {
            if (j < s) red[j] += red[j + s];
            __syncthreads();
        }
        const float norm  = sqrtf(red[0]);
        const float scale = (norm > 0.0f) ? (1.0f / (norm + EPSV)) : 1.0f;
        x_next[j] = h * scale;
    } else if (x_next) {
        x_next[j] = h;
    }
}

// final[r] = sum of KSPLIT partials + b_out[r]
__global__ void out_finish(const float* __restrict__ g,
                           const float* __restrict__ b,
                           float* __restrict__ y)
{
    const int r = blockIdx.x * blockDim.x + threadIdx.x;
    if (r < OUTN) {
        float s = b[r];
#pragma unroll
        for (int p = 0; p < KSPLIT; ++p) s += g[(size_t)p * GH + r];
        y[r] = s;
    }
}

extern "C" void kernel_launch(void* const* d_in, const int* in_sizes, int n_in,
                              void* d_out, int out_size, void* d_ws, size_t ws_size,
                              hipStream_t stream)
{
    const float* x     = (const float*)d_in[0];
    const float* Wih_p = (const float*)d_in[1];
    const float* Whh_p = (const float*)d_in[2];
    const float* bih_p = (const float*)d_in[3];
    const float* bhh_p = (const float*)d_in[4];
    const float* h0_p  = (const float*)d_in[5];
    const float* c0_p  = (const float*)d_in[6];
    const float* Wih_c = (const float*)d_in[7];
    const float* Whh_c = (const float*)d_in[8];
    const float* bih_c = (const float*)d_in[9];
    const float* bhh_c = (const float*)d_in[10];
    const float* h0_c  = (const float*)d_in[11];
    const float* c0_c  = (const float*)d_in[12];
    const float* Wih_h = (const float*)d_in[13];
    const float* Whh_h = (const float*)d_in[14];
    const float* bih_h = (const float*)d_in[15];
    const float* bhh_h = (const float*)d_in[16];
    const float* h0_h  = (const float*)d_in[17];
    const float* c0_h  = (const float*)d_in[18];
    const float* W_out = (const float*)d_in[19];
    const float* b_out = (const float*)d_in[20];

    float* out      = (float*)d_out;
    float* final_o  = out;                               // [512]
    float* h_stack  = out + OUTN;                        // [32*1024]
    float* c_stack  = h_stack + DEPTH * Hn;              // [32*1024]
    float* hp       = c_stack + DEPTH * Hn;              // [1024]
    float* cp       = hp + Hn;
    float* hc       = cp + Hn;
    float* cc       = hc + Hn;

    float* g  = (float*)d_ws;                            // 8*4096 partials
    float* xA = g + 2 * KSPLIT * GH;                     // 1024 current input

    const dim3 blk(WAVES * 32);
    const dim3 grdG(GH / ROWS_PER_BLOCK, 2, KSPLIT);     // (64, 2, 4)
    const dim3 grdO(OUTN / ROWS_PER_BLOCK, 1, KSPLIT);   // (8, 1, 4)

    // pattern LSTM
    gemv_wmma<<<grdG, blk, 0, stream>>>(Wih_p, x, Whh_p, h0_p, g);
    lstm_cell<<<1, Hn, 0, stream>>>(g, bih_p, bhh_p, c0_p, hp, cp, xA, 0);

    // compression LSTM
    gemv_wmma<<<grdG, blk, 0, stream>>>(Wih_c, xA, Whh_c, h0_c, g);
    lstm_cell<<<1, Hn, 0, stream>>>(g, bih_c, bhh_c, c0_c, hc, cc, xA, 0);

    // 32-layer head with inter-layer normalization
    for (int L = 0; L < DEPTH; ++L) {
        gemv_wmma<<<grdG, blk, 0, stream>>>(
            Wih_h + (size_t)L * GH * Hn, xA,
            Whh_h + (size_t)L * GH * Hn, h0_h + (size_t)L * Hn, g);
        lstm_cell<<<1, Hn, 0, stream>>>(
            g, bih_h + (size_t)L * GH, bhh_h + (size_t)L * GH,
            c0_h + (size_t)L * Hn,
            h_stack + (size_t)L * Hn, c_stack + (size_t)L * Hn,
            xA, 1);
    }

    // output linear on raw h of deepest layer
    const float* h_last = h_stack + (size_t)(DEPTH - 1) * Hn;
    gemv_wmma<<<grdO, blk, 0, stream>>>(W_out, h_last, W_out, h_last, g);
    out_finish<<<2, 256, 0, stream>>>(g, b_out, final_o);
}